// AsymmetricTreeLayer_489626271944
// MI455X (gfx1250) — compile-verified
//
#include <hip/hip_runtime.h>
#include <stdint.h>

typedef float v4f __attribute__((ext_vector_type(4)));
typedef float v2f __attribute__((ext_vector_type(2)));

#define ALPHA 0.01f
#define THREADS 256
#define BLOCKS 1024
#define DEPTH 8  // async pipeline depth (LDS buffers); steady-state wait = DEPTH-1

// Issue one 16B/lane async copy global -> LDS (ASYNCcnt-tracked, no dest VGPRs).
// NT hint: input is read exactly once (256 MiB) -- don't pollute the 192 MB L2.
__device__ __forceinline__ void issue_async(unsigned lds_off, const float* gptr) {
    unsigned long long g = (unsigned long long)(uintptr_t)gptr;
    asm volatile("global_load_async_to_lds_b128 %0, %1, off th:TH_LOAD_NT"
                 :: "v"(lds_off), "v"(g) : "memory");
}

// Read the landed chunk from LDS, apply leaky_relu(x*k)+x*cw, pair-sum, NT store.
__device__ __forceinline__ void consume(unsigned lds_off, unsigned c,
                                        v4f k, v4f cw, float* __restrict__ out) {
    v4f x;
    asm volatile("ds_load_b128 %0, %1\n\ts_wait_dscnt 0x0"
                 : "=v"(x) : "v"(lds_off) : "memory");
    float y0 = x.x * k.x; y0 = (y0 > 0.f ? y0 : ALPHA * y0) + x.x * cw.x;
    float y1 = x.y * k.y; y1 = (y1 > 0.f ? y1 : ALPHA * y1) + x.y * cw.y;
    float y2 = x.z * k.z; y2 = (y2 > 0.f ? y2 : ALPHA * y2) + x.z * cw.z;
    float y3 = x.w * k.w; y3 = (y3 > 0.f ? y3 : ALPHA * y3) + x.w * cw.w;
    v2f o;
    o.x = y0 + y1;
    o.y = y2 + y3;
    __builtin_nontemporal_store(o, (v2f*)(out + (size_t)c * 2));
}

__global__ __launch_bounds__(THREADS) void asym_tree_fused_kernel(
    const float* __restrict__ in,    // (BATCH, L) row-major f32
    const float* __restrict__ wk,    // (1, L) f32
    const float* __restrict__ cp,    // (1, L) f32
    float* __restrict__ out,         // (BATCH, L/2) f32
    unsigned nchunks,                // total float4 chunks = BATCH*L/4
    unsigned stride,                 // total threads; multiple of L/4 (pow2)
    int iters,                       // nchunks / stride
    unsigned cgmask)                 // L/4 - 1  (column-group mask, pow2 L)
{
    __shared__ v4f lds[DEPTH][THREADS];  // 32 KB: 8-deep per-thread 16B slots

    const unsigned tid = threadIdx.x;
    const unsigned c0 = blockIdx.x * THREADS + tid;

    // Column group is loop-invariant (stride is a multiple of cgmask+1): one AND.
    const unsigned cg = c0 & cgmask;
    const v4f k  = *(const v4f*)(wk + 4 * cg);
    const v4f cw = *(const v4f*)(cp + 4 * cg);

    unsigned lbuf[DEPTH];
#pragma unroll
    for (int b = 0; b < DEPTH; ++b)
        lbuf[b] = (unsigned)(uintptr_t)(void*)&lds[b][tid];  // LDS byte offset

    // ---- prologue: prime DEPTH-1 chunks ----
#pragma unroll
    for (int j = 0; j < DEPTH - 1; ++j)
        if (j < iters)
            issue_async(lbuf[j], in + ((size_t)c0 + (size_t)j * stride) * 4);

    unsigned c = c0;
    int i = 0;

    // ---- steady state: issue i+DEPTH-1, wait fixed DEPTH-1, consume i ----
    for (; i + (DEPTH - 1) < iters; ++i) {
        issue_async(lbuf[(i + DEPTH - 1) & (DEPTH - 1)],
                    in + ((size_t)c + (size_t)(DEPTH - 1) * stride) * 4);
        asm volatile("s_wait_asynccnt 0x7" ::: "memory");
        consume(lbuf[i & (DEPTH - 1)], c, k, cw, out);
        c += stride;
    }

    // ---- drain: at most DEPTH-1 iterations, wait = chunks still outstanding ----
    for (; i < iters; ++i) {
        switch (iters - 1 - i) {
            case 0:  asm volatile("s_wait_asynccnt 0x0" ::: "memory"); break;
            case 1:  asm volatile("s_wait_asynccnt 0x1" ::: "memory"); break;
            case 2:  asm volatile("s_wait_asynccnt 0x2" ::: "memory"); break;
            case 3:  asm volatile("s_wait_asynccnt 0x3" ::: "memory"); break;
            case 4:  asm volatile("s_wait_asynccnt 0x4" ::: "memory"); break;
            case 5:  asm volatile("s_wait_asynccnt 0x5" ::: "memory"); break;
            default: asm volatile("s_wait_asynccnt 0x6" ::: "memory"); break;
        }
        consume(lbuf[i & (DEPTH - 1)], c, k, cw, out);
        c += stride;
    }

    // ---- tail (general sizes; empty for the reference shape) ----
    for (unsigned ct = (unsigned)iters * stride + c0; ct < nchunks; ct += stride) {
        v4f x = __builtin_nontemporal_load((const v4f*)(in + (size_t)ct * 4));
        float y0 = x.x * k.x; y0 = (y0 > 0.f ? y0 : ALPHA * y0) + x.x * cw.x;
        float y1 = x.y * k.y; y1 = (y1 > 0.f ? y1 : ALPHA * y1) + x.y * cw.y;
        float y2 = x.z * k.z; y2 = (y2 > 0.f ? y2 : ALPHA * y2) + x.z * cw.z;
        float y3 = x.w * k.w; y3 = (y3 > 0.f ? y3 : ALPHA * y3) + x.w * cw.w;
        v2f o; o.x = y0 + y1; o.y = y2 + y3;
        __builtin_nontemporal_store(o, (v2f*)(out + (size_t)ct * 2));
    }
}

extern "C" void kernel_launch(void* const* d_in, const int* in_sizes, int n_in,
                              void* d_out, int out_size, void* d_ws, size_t ws_size,
                              hipStream_t stream) {
    (void)n_in; (void)out_size; (void)d_ws; (void)ws_size;

    const float* in = (const float*)d_in[0];   // (BATCH, L) f32
    const float* wk = (const float*)d_in[1];   // (1, L) f32
    const float* cp = (const float*)d_in[2];   // (1, L) f32
    float* out = (float*)d_out;                // (BATCH, L/2) f32

    const unsigned n0 = (unsigned)in_sizes[0];     // BATCH*L = 2^26
    const unsigned L  = (unsigned)in_sizes[1];     // 256 (power of two)
    const unsigned cgmask = L / 4 - 1;             // 63

    const unsigned nchunks = n0 / 4;               // 2^24 float4 chunks
    const unsigned stride  = BLOCKS * THREADS;     // 2^18, multiple of L/4
    const int      iters   = (int)(nchunks / stride);  // 64

    asym_tree_fused_kernel<<<BLOCKS, THREADS, 0, stream>>>(
        in, wk, cp, out, nchunks, stride, iters, cgmask);
}